// NonSpikingLayer_18537078849673
// MI455X (gfx1250) — compile-verified
//
#include <hip/hip_runtime.h>
#include <stdint.h>

// Problem constants (match reference).
#define NN 128
#define TT 2000
#define TC 40                      // timesteps per chunk (2000 = 50 * 40), 160B/thread
#define NCHUNK (TT / TC)           // 50
#define CPB 64                     // cells per block (half a row -> i is block-uniform)

typedef __attribute__((ext_vector_type(4))) unsigned int v4u;
typedef __attribute__((ext_vector_type(8))) int          v8i;
typedef __attribute__((ext_vector_type(4))) int          v4i;
typedef __attribute__((ext_vector_type(4))) float        v4f;

__global__ __launch_bounds__(64) void nonspiking_rk4_tdm(
    const float* __restrict__ up0,   // [N,T]
    const float* __restrict__ up1,   // [N,T]
    const float* __restrict__ ks,    // [2,N]
    float* __restrict__ out)         // [N,N,T]
{
  __shared__ __align__(16) float lds[2][CPB * TC];   // 2 x 10 KB double buffer

  const int local = threadIdx.x;                 // 0..63
  const int cell0 = blockIdx.x * CPB;
  // i is provably uniform per block: 64 consecutive cells never cross a row.
  const int i = blockIdx.x >> 1;                 // row (uniform -> scalar loads for up0)
  const int j = ((blockIdx.x & 1) << 6) + local; // col

  // g_max = (k*R)/(E_SYN_DELTA - k*R), R=1, E_SYN_DELTA=20
  const float k0 = ks[i];
  const float k1 = ks[NN + j];
  const float gm0 = k0 / (20.0f - k0);
  const float gm1 = k1 / (20.0f - k1);

  const float* __restrict__ row0 = up0 + i * TT; // uniform base
  const float* __restrict__ row1 = up1 + j * TT;

  float v = 0.0f;   // E_REST

  const uint64_t gbase = (uint64_t)(uintptr_t)out + (uint64_t)cell0 * TT * 4ull;

  for (int c = 0; c < NCHUNK; ++c) {
    float vbuf[TC];
    const int tbase = c * TC;

#pragma unroll
    for (int tt = 0; tt < TC; ++tt) {
      const int t = tbase + tt;
      float u0 = row0[t];                        // uniform address -> SMEM path
      float u1 = row1[t];
      float a0 = fminf(fmaxf(u0, 0.0f), 1.0f);
      float a1 = fminf(fmaxf(u1, 0.0f), 1.0f);
      float g    = gm0 * a0 + gm1 * a1;
      float isyn = g * (20.0f - v);              // frozen over RK4 substeps
      // dv(vv) = isyn - 5*vv ; DT = 0.1
      float r1 = isyn - 5.0f * v;
      float r2 = isyn - 5.0f * (v + 0.05f * r1);
      float r3 = isyn - 5.0f * (v + 0.05f * r2);
      float r4 = isyn - 5.0f * (v + 0.10f * r3);
      v = v + (0.1f / 6.0f) * (r1 + 2.0f * r2 + 2.0f * r3 + r4);
      vbuf[tt] = v;
    }

    const int p = c & 1;

    // Before reusing LDS buffer p (read by chunk c-2's TDM store), make
    // sure at most the chunk c-1 store is still in flight.
    if (threadIdx.x < 32) {
      __builtin_amdgcn_s_wait_tensorcnt(1);
    }
    __syncthreads();

    // Stage this thread's 40 contiguous outputs in LDS (10 x ds_store_b128).
    float* dst = &lds[p][local * TC];
#pragma unroll
    for (int q = 0; q < TC / 4; ++q) {
      v4f val = { vbuf[4*q+0], vbuf[4*q+1], vbuf[4*q+2], vbuf[4*q+3] };
      *(v4f*)(dst + 4 * q) = val;
    }
    __syncthreads();

    // Wave 0 launches the TDM: LDS tile (CPB rows x TC elems, contiguous)
    // -> global 2D tile: row y at gaddr + 4*(x + y*2000).
    if (threadIdx.x < 32) {
      const uint32_t lds_off = (uint32_t)(uintptr_t)&lds[p][0]; // low 32b = LDS byte offset
      const uint64_t gaddr   = gbase + (uint64_t)tbase * 4ull;

      v4u g0d;
      g0d.x = 1u;                                   // count=1 (valid), user mode
      g0d.y = lds_off;                              // lds_addr
      g0d.z = (uint32_t)gaddr;                      // global_addr[31:0]
      g0d.w = (uint32_t)((gaddr >> 32) & 0x1FFFFFFu) | (2u << 30); // addr[56:32] | type=2

      v8i g1d;
      g1d[0] = (int)(2u << 16);                     // workgroup_mask=0, data_size=2 (4B)
      g1d[1] = (int)(((uint32_t)TT & 0xFFFFu) << 16);               // tensor_dim0[15:0]
      g1d[2] = (int)((((uint32_t)TT >> 16) & 0xFFFFu)               // tensor_dim0[31:16]
                     | (((uint32_t)(NN * NN)) << 16));              // tensor_dim1[15:0]
      g1d[3] = (int)(((uint32_t)TC) << 16);         // tensor_dim1[31:16]=0 | tile_dim0=TC
      g1d[4] = (int)CPB;                            // tile_dim1=64, tile_dim2=0
      g1d[5] = (int)TT;                             // tensor_dim0_stride[31:0] = 2000
      g1d[6] = 0;                                   // stride hi | tensor_dim1_stride lo
      g1d[7] = 0;

      v4i zero4 = {0, 0, 0, 0};
      v8i zero8 = {0, 0, 0, 0, 0, 0, 0, 0};
      // 6-arg clang-23 form: (g0, g1, g2, g3, extra, cpol). 2D tensor -> groups 2/3 unused.
      __builtin_amdgcn_tensor_store_from_lds(g0d, g1d, zero4, zero4, zero8, 0);
    }
  }

  if (threadIdx.x < 32) {
    __builtin_amdgcn_s_wait_tensorcnt(0);
  }
}

extern "C" void kernel_launch(void* const* d_in, const int* in_sizes, int n_in,
                              void* d_out, int out_size, void* d_ws, size_t ws_size,
                              hipStream_t stream) {
  const float* up0 = (const float*)d_in[0];   // u_pre_0 [128,2000]
  const float* up1 = (const float*)d_in[1];   // u_pre_1 [128,2000]
  const float* ks  = (const float*)d_in[2];   // k_syn   [2,128]
  float* out = (float*)d_out;                 // [128,128,2000]

  dim3 grid(NN * NN / CPB);   // 256 blocks -> spread across WGPs / TDM engines
  dim3 block(CPB);            // 64 threads = 2 waves
  nonspiking_rk4_tdm<<<grid, block, 0, stream>>>(up0, up1, ks, out);
}